// QuantizationLayer_43396349559001
// MI455X (gfx1250) — compile-verified
//
#include <hip/hip_runtime.h>

// ---------------------------------------------------------------------------
// Order-preserving float <-> uint32 key mapping (monotonic), so global
// min/max combining is a plain u32 atomicMin/atomicMax (no CAS loop).
// ---------------------------------------------------------------------------
static __device__ __forceinline__ unsigned fkey(float f) {
    unsigned u = __float_as_uint(f);
    return (u & 0x80000000u) ? ~u : (u | 0x80000000u);
}
static __device__ __forceinline__ float funkey(unsigned k) {
    unsigned u = (k & 0x80000000u) ? (k & 0x7fffffffu) : ~k;
    return __uint_as_float(u);
}

static __device__ __forceinline__ float f4min(float4 v) {
    return fminf(fminf(v.x, v.y), fminf(v.z, v.w));
}
static __device__ __forceinline__ float f4max(float4 v) {
    return fmaxf(fmaxf(v.x, v.y), fmaxf(v.z, v.w));
}

// ws[0] = running MIN key (init to max key), ws[1] = running MAX key (init 0)
__global__ void qz_init_minmax(unsigned* __restrict__ ws) {
    ws[0] = 0xFFFFFFFFu;
    ws[1] = 0x00000000u;
}

// ---------------------------------------------------------------------------
// Pass 1: global min/max reduction. 4x-unrolled B128 streaming loads (4
// outstanding loads per wave before the wait) + speculative global_prefetch_b8;
// wave32 butterfly reduction; per-wave LDS u32 atomics; 2 device atomics/block.
// 32-bit indexing throughout (nvec < 2^31).
// ---------------------------------------------------------------------------
__global__ void qz_reduce_minmax(const float4* __restrict__ x, int nvec,
                                 const float* __restrict__ tail, int ntail,
                                 unsigned* __restrict__ ws) {
    float vmin =  __builtin_huge_valf();
    float vmax = -__builtin_huge_valf();

    const int stride = (int)(gridDim.x * blockDim.x);
    int i = (int)(blockIdx.x * blockDim.x + threadIdx.x);

    // main loop: 4 independent B128 loads per iteration
    for (; i + 3 * stride < nvec; i += 4 * stride) {
        __builtin_prefetch((const void*)(x + i + 4 * stride), 0, 0); // speculative
        float4 a = x[i];
        float4 b = x[i + stride];
        float4 c = x[i + 2 * stride];
        float4 d = x[i + 3 * stride];
        vmin = fminf(vmin, fminf(fminf(f4min(a), f4min(b)), fminf(f4min(c), f4min(d))));
        vmax = fmaxf(vmax, fmaxf(fmaxf(f4max(a), f4max(b)), fmaxf(f4max(c), f4max(d))));
    }
    for (; i < nvec; i += stride) {
        float4 v = x[i];
        vmin = fminf(vmin, f4min(v));
        vmax = fmaxf(vmax, f4max(v));
    }
    if (blockIdx.x == 0 && (int)threadIdx.x < ntail) {
        float t = tail[threadIdx.x];
        vmin = fminf(vmin, t);
        vmax = fmaxf(vmax, t);
    }

    // wave32 butterfly reduction
    #pragma unroll
    for (int off = 16; off > 0; off >>= 1) {
        vmin = fminf(vmin, __shfl_xor(vmin, off, 32));
        vmax = fmaxf(vmax, __shfl_xor(vmax, off, 32));
    }

    __shared__ unsigned s[2];
    if (threadIdx.x == 0) { s[0] = 0xFFFFFFFFu; s[1] = 0u; }
    __syncthreads();
    if ((threadIdx.x & 31) == 0) {            // lane 0 of each wave (wave32)
        atomicMin(&s[0], fkey(vmin));         // ds_min_u32
        atomicMax(&s[1], fkey(vmax));         // ds_max_u32
    }
    __syncthreads();
    if (threadIdx.x == 0) {
        atomicMin(&ws[0], s[0]);              // global_atomic_min_u32
        atomicMax(&ws[1], s[1]);              // global_atomic_max_u32
    }
}

// ---------------------------------------------------------------------------
// Pass 2: snap each element to nearest uniform center.
// center = xmin + (clamp(floor((x - xmin) * L/(xmax-xmin)), 0, L-1) + 0.5)*step
// 4x-unrolled B128 load/store; reads largely hit the 192MB L2 warmed by pass 1.
// ---------------------------------------------------------------------------
static __device__ __forceinline__ float qsnap(float v, float xmin, float inv,
                                              float lm1, float step) {
    return __fmaf_rn(fminf(fmaxf(floorf((v - xmin) * inv), 0.f), lm1) + 0.5f,
                     step, xmin);
}
static __device__ __forceinline__ float4 qsnap4(float4 v, float xmin, float inv,
                                                float lm1, float step) {
    float4 q;
    q.x = qsnap(v.x, xmin, inv, lm1, step);
    q.y = qsnap(v.y, xmin, inv, lm1, step);
    q.z = qsnap(v.z, xmin, inv, lm1, step);
    q.w = qsnap(v.w, xmin, inv, lm1, step);
    return q;
}

__global__ void qz_quantize(const float4* __restrict__ x, float4* __restrict__ out,
                            int nvec,
                            const float* __restrict__ xt, float* __restrict__ ot,
                            int ntail,
                            const unsigned* __restrict__ ws,
                            const int* __restrict__ nbits_p) {
    const float xmin = funkey(ws[0]);
    const float xmax = funkey(ws[1]);
    const int   levels = 1 << nbits_p[0];
    const float step = (xmax - xmin) / (float)levels;
    const float inv  = (float)levels / (xmax - xmin);
    const float lm1  = (float)(levels - 1);

    const int stride = (int)(gridDim.x * blockDim.x);
    int i = (int)(blockIdx.x * blockDim.x + threadIdx.x);

    for (; i + 3 * stride < nvec; i += 4 * stride) {
        __builtin_prefetch((const void*)(x + i + 4 * stride), 0, 0);
        float4 a = x[i];
        float4 b = x[i + stride];
        float4 c = x[i + 2 * stride];
        float4 d = x[i + 3 * stride];
        out[i]              = qsnap4(a, xmin, inv, lm1, step);
        out[i + stride]     = qsnap4(b, xmin, inv, lm1, step);
        out[i + 2 * stride] = qsnap4(c, xmin, inv, lm1, step);
        out[i + 3 * stride] = qsnap4(d, xmin, inv, lm1, step);
    }
    for (; i < nvec; i += stride) {
        out[i] = qsnap4(x[i], xmin, inv, lm1, step);
    }
    if (blockIdx.x == 0 && (int)threadIdx.x < ntail) {
        ot[threadIdx.x] = qsnap(xt[threadIdx.x], xmin, inv, lm1, step);
    }
}

// ---------------------------------------------------------------------------
// Host-side launcher (graph-capture safe: only kernel launches on `stream`).
// d_in[0] = x (f32, N elems), d_in[1] = n_bits (int, 1 elem).
// d_out   = N f32.  d_ws[0..1] = u32 min/max keys (re-seeded every call).
// ---------------------------------------------------------------------------
extern "C" void kernel_launch(void* const* d_in, const int* in_sizes, int n_in,
                              void* d_out, int out_size, void* d_ws, size_t ws_size,
                              hipStream_t stream) {
    const float* x     = (const float*)d_in[0];
    const int*   nbits = (const int*)d_in[1];
    float*       out   = (float*)d_out;
    unsigned*    ws    = (unsigned*)d_ws;

    const long N    = (long)in_sizes[0];
    const long nvec = N >> 2;           // float4 count (fits in int32 here)
    const int  rem  = (int)(N & 3);     // scalar tail
    const float* tail_in  = x + (nvec << 2);
    float*       tail_out = out + (nvec << 2);

    const int threads = 256;            // 8 waves (wave32) per workgroup
    // Size grid so each thread covers ~4 float4s in one unrolled pass.
    long blk = (nvec + (long)threads * 4 - 1) / ((long)threads * 4);
    if (blk < 1) blk = 1;
    if (blk > 12288) blk = 12288;       // 12288*256*4 = 12,582,912 = nvec @ full size
    const int blocks = (int)blk;

    qz_init_minmax<<<1, 1, 0, stream>>>(ws);
    qz_reduce_minmax<<<blocks, threads, 0, stream>>>(
        (const float4*)x, (int)nvec, tail_in, rem, ws);
    qz_quantize<<<blocks, threads, 0, stream>>>(
        (const float4*)x, (float4*)out, (int)nvec, tail_in, tail_out, rem, ws, nbits);
}